// GatNet_7576322310703
// MI455X (gfx1250) — compile-verified
//
#include <hip/hip_runtime.h>
#include <math.h>

typedef __attribute__((ext_vector_type(2))) float v2f;
typedef __attribute__((ext_vector_type(8))) float v8f;

#define NEG_SLOPE 0.2f
#define SM_EPS 1e-16f

__device__ __forceinline__ float leakyf(float v) { return v > 0.f ? v : NEG_SLOPE * v; }
__device__ __forceinline__ float eluf(float v)   { return v > 0.f ? v : (__expf(v) - 1.f); }

// Sign-aware float atomic max: works given buffer initialized to -inf.
__device__ __forceinline__ void atomicMaxF(float* addr, float v) {
  if (v >= 0.f) atomicMax((int*)addr, __float_as_int(v));
  else          atomicMin((unsigned int*)addr, __float_as_uint(v));
}

// ---------------- Layer 1: x[N,1] * W1[1,32] outer product + att dots + init ----
__global__ void k_l1_transform(const float* __restrict__ x, const float* __restrict__ W1,
                               const float* __restrict__ att_s, const float* __restrict__ att_d,
                               float* __restrict__ h1, float* __restrict__ as_n,
                               float* __restrict__ ad_n, float* __restrict__ mbuf,
                               float* __restrict__ sbuf, float* __restrict__ outbuf, int N) {
  int n = blockIdx.x * blockDim.x + threadIdx.x;
  if (n >= N) return;
  float xv = x[n];
  float as[4] = {0.f,0.f,0.f,0.f}, ad[4] = {0.f,0.f,0.f,0.f};
  #pragma unroll
  for (int j = 0; j < 32; ++j) {
    float hv = xv * W1[j];
    h1[(size_t)n*32 + j] = hv;
    outbuf[(size_t)n*32 + j] = 0.f;
    as[j >> 3] += hv * att_s[j];
    ad[j >> 3] += hv * att_d[j];
  }
  #pragma unroll
  for (int h = 0; h < 4; ++h) {
    as_n[(size_t)n*4 + h] = as[h];
    ad_n[(size_t)n*4 + h] = ad[h];
    mbuf[(size_t)n*4 + h] = -__builtin_inff();
    sbuf[(size_t)n*4 + h] = 0.f;
  }
}

// ---------------- Edge pass 1: segment max of leaky-relu score -------------------
template<int H>
__global__ void k_edge_max(const int* __restrict__ esrc, const int* __restrict__ edst,
                           int E, int N,
                           const float* __restrict__ as_n, const float* __restrict__ ad_n,
                           float* __restrict__ mbuf) {
  int e = blockIdx.x * blockDim.x + threadIdx.x;
  if (e >= E + N) return;
  int s = (e < E) ? esrc[e] : (e - E);   // implicit self loops appended
  int d = (e < E) ? edst[e] : (e - E);
  #pragma unroll
  for (int h = 0; h < H; ++h) {
    float sc = leakyf(as_n[(size_t)s*H + h] + ad_n[(size_t)d*H + h]);
    atomicMaxF(&mbuf[(size_t)d*H + h], sc);
  }
}

// ---------------- Edge pass 2: segment sum of exp(score - max) -------------------
template<int H>
__global__ void k_edge_expsum(const int* __restrict__ esrc, const int* __restrict__ edst,
                              int E, int N,
                              const float* __restrict__ as_n, const float* __restrict__ ad_n,
                              const float* __restrict__ mbuf, float* __restrict__ sbuf) {
  int e = blockIdx.x * blockDim.x + threadIdx.x;
  if (e >= E + N) return;
  int s = (e < E) ? esrc[e] : (e - E);
  int d = (e < E) ? edst[e] : (e - E);
  #pragma unroll
  for (int h = 0; h < H; ++h) {
    float sc = leakyf(as_n[(size_t)s*H + h] + ad_n[(size_t)d*H + h]);
    atomicAdd(&sbuf[(size_t)d*H + h], __expf(sc - mbuf[(size_t)d*H + h]));
  }
}

// ---------------- Edge pass 3: scatter alpha * h[src] into out[dst] --------------
template<int H, int C>
__global__ void k_edge_scatter(const int* __restrict__ esrc, const int* __restrict__ edst,
                               int E, int N,
                               const float* __restrict__ as_n, const float* __restrict__ ad_n,
                               const float* __restrict__ mbuf, const float* __restrict__ sbuf,
                               const float* __restrict__ hfeat, float* __restrict__ outbuf) {
  int e = blockIdx.x * blockDim.x + threadIdx.x;
  if (e >= E + N) return;
  int s = (e < E) ? esrc[e] : (e - E);
  int d = (e < E) ? edst[e] : (e - E);
  // Kick the h[src] gather rows toward the caches while scores/alpha compute.
  __builtin_prefetch(hfeat + (size_t)s * H * C, 0, 3);
  __builtin_prefetch(hfeat + (size_t)s * H * C + (H * C / 2), 0, 3);
  #pragma unroll
  for (int h = 0; h < H; ++h) {
    float sc = leakyf(as_n[(size_t)s*H + h] + ad_n[(size_t)d*H + h]);
    float alpha = __expf(sc - mbuf[(size_t)d*H + h]) / (sbuf[(size_t)d*H + h] + SM_EPS);
    const float* hs = hfeat + ((size_t)s*H + h) * C;
    float* od = outbuf + ((size_t)d*H + h) * C;
    #pragma unroll
    for (int c = 0; c < C; ++c) atomicAdd(&od[c], alpha * hs[c]);
  }
}

// ---------------- elementwise bias + ELU ----------------------------------------
__global__ void k_bias_elu(const float* __restrict__ in, const float* __restrict__ bias,
                           float* __restrict__ out, int total, int F) {
  int i = blockIdx.x * blockDim.x + threadIdx.x;
  if (i >= total) return;
  out[i] = eluf(in[i] + bias[i % F]);
}

// ---------------- WMMA GEMM: A[N,32] x B[32,64] -> C[N,64] (f32) -----------------
// W2 staged in LDS with rows (2k, 2k+1) column-interleaved so each lane's B
// operand v2f{B[k][c], B[k+1][c]} is one aligned ds_load_b64 into an even VGPR
// pair (no repacking movs). K-pair stride 160 floats (128 data + 32 pad):
//   half 0 reads banks {2m, 2m+1} (0..31), half 1 reads {32+2m, 33+2m} (32..63)
// -> all 64 bank requests distinct, conflict-free.
// One wave computes a 16x64 strip with V_WMMA_F32_16X16X4_F32 (ISA 7.12.2
// operand layouts: lanes 0-15 hold K={0,1}, lanes 16-31 hold K={2,3}).
#define KP_STRIDE 160
__global__ void k_gemm_wmma(const float* __restrict__ A, const float* __restrict__ B,
                            float* __restrict__ C, int Nrows) {
  __shared__ float Bs[16 * KP_STRIDE];   // 10 KB
  for (int j = threadIdx.x; j < 16 * 64; j += blockDim.x) {
    int kp = j >> 6, c = j & 63;
    Bs[kp * KP_STRIDE + 2*c + 0] = B[(2*kp + 0) * 64 + c];
    Bs[kp * KP_STRIDE + 2*c + 1] = B[(2*kp + 1) * 64 + c];
  }
  __syncthreads();

  int wave = threadIdx.x >> 5;           // 4 waves per block
  int lane = threadIdx.x & 31;
  int row0 = (blockIdx.x * 4 + wave) * 16;
  if (row0 + 16 <= Nrows) {              // Nrows is a multiple of 16 here
    int half = lane >> 4;                // 0 or 1
    int m = lane & 15;

    v8f acc0 = {}, acc1 = {}, acc2 = {}, acc3 = {};
    #pragma unroll
    for (int kk = 0; kk < 8; ++kk) {
      int kbase = kk * 4 + half * 2;     // even -> 8B aligned v2f load
      v2f a = *(const v2f*)(A + (size_t)(row0 + m) * 32 + kbase);
      const v2f* brow = (const v2f*)(Bs + (2*kk + half) * KP_STRIDE);
      v2f b0 = brow[ 0 + m];
      v2f b1 = brow[16 + m];
      v2f b2 = brow[32 + m];
      v2f b3 = brow[48 + m];
      acc0 = __builtin_amdgcn_wmma_f32_16x16x4_f32(false, a, false, b0, (short)0, acc0, false, false);
      acc1 = __builtin_amdgcn_wmma_f32_16x16x4_f32(false, a, false, b1, (short)0, acc1, false, false);
      acc2 = __builtin_amdgcn_wmma_f32_16x16x4_f32(false, a, false, b2, (short)0, acc2, false, false);
      acc3 = __builtin_amdgcn_wmma_f32_16x16x4_f32(false, a, false, b3, (short)0, acc3, false, false);
    }
    #pragma unroll
    for (int i = 0; i < 8; ++i) {
      size_t row = (size_t)(row0 + i + half * 8);
      C[row * 64 +  0 + m] = acc0[i];
      C[row * 64 + 16 + m] = acc1[i];
      C[row * 64 + 32 + m] = acc2[i];
      C[row * 64 + 48 + m] = acc3[i];
    }
  }
}

// ---------------- Layer 2 attention dots + buffer init ---------------------------
__global__ void k_l2_atts(const float* __restrict__ h2,
                          const float* __restrict__ att_s, const float* __restrict__ att_d,
                          float* __restrict__ as_n, float* __restrict__ ad_n,
                          float* __restrict__ mbuf, float* __restrict__ sbuf,
                          float* __restrict__ outbuf, int N) {
  int n = blockIdx.x * blockDim.x + threadIdx.x;
  if (n >= N) return;
  float as[8] = {}, ad[8] = {};
  #pragma unroll
  for (int j = 0; j < 64; ++j) {
    float hv = h2[(size_t)n*64 + j];
    outbuf[(size_t)n*64 + j] = 0.f;
    as[j >> 3] += hv * att_s[j];
    ad[j >> 3] += hv * att_d[j];
  }
  #pragma unroll
  for (int h = 0; h < 8; ++h) {
    as_n[(size_t)n*8 + h] = as[h];
    ad_n[(size_t)n*8 + h] = ad[h];
    mbuf[(size_t)n*8 + h] = -__builtin_inff();
    sbuf[(size_t)n*8 + h] = 0.f;
  }
}

// ---------------- Final: bias + ELU + fc (64 -> 1) -------------------------------
__global__ void k_final(const float* __restrict__ out2, const float* __restrict__ bias2,
                        const float* __restrict__ fcw, const float* __restrict__ fcb,
                        float* __restrict__ y, int N) {
  int n = blockIdx.x * blockDim.x + threadIdx.x;
  if (n >= N) return;
  float acc = 0.f;
  #pragma unroll
  for (int j = 0; j < 64; ++j)
    acc += eluf(out2[(size_t)n*64 + j] + bias2[j]) * fcw[j];
  y[n] = acc + fcb[0];
}

extern "C" void kernel_launch(void* const* d_in, const int* in_sizes, int n_in,
                              void* d_out, int out_size, void* d_ws, size_t ws_size,
                              hipStream_t stream) {
  const float* x    = (const float*)d_in[0];
  const int*   eidx = (const int*)  d_in[1];
  const float* W1   = (const float*)d_in[2];
  const float* as1  = (const float*)d_in[3];
  const float* ad1  = (const float*)d_in[4];
  const float* b1   = (const float*)d_in[5];
  const float* W2   = (const float*)d_in[6];
  const float* as2  = (const float*)d_in[7];
  const float* ad2  = (const float*)d_in[8];
  const float* b2   = (const float*)d_in[9];
  const float* fcw  = (const float*)d_in[10];
  const float* fcb  = (const float*)d_in[11];

  int N = in_sizes[0];          // x is [N,1]
  int E = in_sizes[1] / 2;      // edge_index is [2,E]
  const int* esrc = eidx;
  const int* edst = eidx + E;

  float* ws = (float*)d_ws;
  size_t o = 0;
  float* h1   = ws + o; o += (size_t)N * 32;
  float* as1n = ws + o; o += (size_t)N * 4;
  float* ad1n = ws + o; o += (size_t)N * 4;
  float* m1   = ws + o; o += (size_t)N * 4;
  float* s1   = ws + o; o += (size_t)N * 4;
  float* out1 = ws + o; o += (size_t)N * 32;
  float* h1a  = ws + o; o += (size_t)N * 32;
  float* h2   = ws + o; o += (size_t)N * 64;
  float* as2n = ws + o; o += (size_t)N * 8;
  float* ad2n = ws + o; o += (size_t)N * 8;
  float* m2   = ws + o; o += (size_t)N * 8;
  float* s2   = ws + o; o += (size_t)N * 8;
  float* out2 = ws + o; o += (size_t)N * 64;

  const int TB = 256;
  int gN = (N + TB - 1) / TB;
  int gE = (E + N + TB - 1) / TB;
  int gElem1 = ((N * 32) + TB - 1) / TB;

  // ---- Layer 1 ----
  k_l1_transform<<<gN, TB, 0, stream>>>(x, W1, as1, ad1, h1, as1n, ad1n, m1, s1, out1, N);
  k_edge_max<4>   <<<gE, TB, 0, stream>>>(esrc, edst, E, N, as1n, ad1n, m1);
  k_edge_expsum<4><<<gE, TB, 0, stream>>>(esrc, edst, E, N, as1n, ad1n, m1, s1);
  k_edge_scatter<4,8><<<gE, TB, 0, stream>>>(esrc, edst, E, N, as1n, ad1n, m1, s1, h1, out1);
  k_bias_elu<<<gElem1, TB, 0, stream>>>(out1, b1, h1a, N * 32, 32);

  // ---- Layer 2 dense transform via WMMA (LDS-staged, pair-packed B) ----
  int gGemm = (N + 63) / 64;    // 4 waves/block * 16 rows/wave
  k_gemm_wmma<<<gGemm, 128, 0, stream>>>(h1a, W2, h2, N);

  // ---- Layer 2 attention ----
  k_l2_atts<<<gN, TB, 0, stream>>>(h2, as2, ad2, as2n, ad2n, m2, s2, out2, N);
  k_edge_max<8>   <<<gE, TB, 0, stream>>>(esrc, edst, E, N, as2n, ad2n, m2);
  k_edge_expsum<8><<<gE, TB, 0, stream>>>(esrc, edst, E, N, as2n, ad2n, m2, s2);
  k_edge_scatter<8,8><<<gE, TB, 0, stream>>>(esrc, edst, E, N, as2n, ad2n, m2, s2, h2, out2);

  // ---- Final bias+ELU+FC ----
  k_final<<<gN, TB, 0, stream>>>(out2, b2, fcw, fcb, (float*)d_out, N);
}